// NestedEventMamba_21921513079115
// MI455X (gfx1250) — compile-verified
//
#include <hip/hip_runtime.h>
#include <math.h>

// Model dims (fixed by reference)
#define SEGS 64
#define LMAXC 256
#define HD 128
#define FEATC 130
#define DIC 256
#define DSC 16
#define SPC 8
#define BATCH 8
#define NCC 11

typedef float v2f __attribute__((ext_vector_type(2)));
typedef float v8f __attribute__((ext_vector_type(8)));

__device__ __forceinline__ float apply_act(float x, int act) {
  switch (act) {
    case 1: return 0.5f * x * (1.0f + erff(x * 0.70710678118654752440f)); // exact GELU
    case 2: return x / (1.0f + expf(-x));                                 // SiLU
    case 3: return (x > 20.0f) ? x : log1pf(expf(x));                     // softplus
    default: return x;
  }
}

// ---------------- fast GEMM: no guards, 16x64 tile per wave32 ----------------
// Requires K%4==0, M%16==0 (grid.x=M/16), O%64==0 (grid.y=O/64).
// Software-pipelined: loads for chunk k+4 are issued before the 4 WMMAs of
// chunk k, so s_wait_loadcnt overlaps with matrix work.
// Y[m,o] = act(sum_k X[m,k]*W[o,k] + bias[o]) + res[m,o]
__global__ void gemm_wmma_f32_fast(const float* __restrict__ X, int xs,
                                   const float* __restrict__ W,
                                   const float* bias, const float* res,
                                   float* Y, int K, int O, int act) {
  const int l = threadIdx.x;
  const int mr = l & 15;
  const int half = l >> 4;
  const int tileM = blockIdx.x << 4;
  const int tileO = blockIdx.y << 6;
  const float* xp = X + (size_t)(tileM + mr) * xs + (half << 1);
  const float* wp = W + (size_t)(tileO + mr) * K + (half << 1);
  const size_t ws1 = (size_t)16 * K;
  v8f acc0 = {0.f, 0.f, 0.f, 0.f, 0.f, 0.f, 0.f, 0.f};
  v8f acc1 = acc0, acc2 = acc0, acc3 = acc0;

  v2f a  = *(const v2f*)(xp);
  v2f b0 = *(const v2f*)(wp);
  v2f b1 = *(const v2f*)(wp + ws1);
  v2f b2 = *(const v2f*)(wp + 2 * ws1);
  v2f b3 = *(const v2f*)(wp + 3 * ws1);
#pragma unroll 4
  for (int k0 = 4; k0 < K; k0 += 4) {
    const v2f an  = *(const v2f*)(xp + k0);
    const v2f bn0 = *(const v2f*)(wp + k0);
    const v2f bn1 = *(const v2f*)(wp + ws1 + k0);
    const v2f bn2 = *(const v2f*)(wp + 2 * ws1 + k0);
    const v2f bn3 = *(const v2f*)(wp + 3 * ws1 + k0);
    acc0 = __builtin_amdgcn_wmma_f32_16x16x4_f32(false, a, false, b0, (short)0, acc0, false, false);
    acc1 = __builtin_amdgcn_wmma_f32_16x16x4_f32(false, a, false, b1, (short)0, acc1, false, false);
    acc2 = __builtin_amdgcn_wmma_f32_16x16x4_f32(false, a, false, b2, (short)0, acc2, false, false);
    acc3 = __builtin_amdgcn_wmma_f32_16x16x4_f32(false, a, false, b3, (short)0, acc3, false, false);
    a = an; b0 = bn0; b1 = bn1; b2 = bn2; b3 = bn3;
  }
  acc0 = __builtin_amdgcn_wmma_f32_16x16x4_f32(false, a, false, b0, (short)0, acc0, false, false);
  acc1 = __builtin_amdgcn_wmma_f32_16x16x4_f32(false, a, false, b1, (short)0, acc1, false, false);
  acc2 = __builtin_amdgcn_wmma_f32_16x16x4_f32(false, a, false, b2, (short)0, acc2, false, false);
  acc3 = __builtin_amdgcn_wmma_f32_16x16x4_f32(false, a, false, b3, (short)0, acc3, false, false);

  v8f accs[4] = {acc0, acc1, acc2, acc3};
#pragma unroll
  for (int j = 0; j < 4; ++j) {
    const int col = tileO + (j << 4) + mr;
    const float bb = bias ? bias[col] : 0.0f;
#pragma unroll
    for (int v = 0; v < 8; ++v) {
      const int row = tileM + v + (half << 3);
      float val = apply_act(accs[j][v] + bb, act);
      if (res) val += res[(size_t)row * O + col];
      Y[(size_t)row * O + col] = val;
    }
  }
}

// ---------------- generic GEMM: edge tiles, branch-free guarded loads -------
__global__ void gemm_wmma_f32_gen(const float* X, int xs, const float* W,
                                  const float* bias, const float* res, float* Y,
                                  int M, int K, int O, int act) {
  const int l = threadIdx.x;
  const int mr = l & 15;
  const int half = l >> 4;
  const int tileM = blockIdx.x << 4;
  const int tileO = blockIdx.y << 4;
  // clamp indices, zero via multiplicative mask -> unconditional loads
  const int arow = min(tileM + mr, M - 1);
  const float am = (tileM + mr < M) ? 1.0f : 0.0f;
  const int bcol = min(tileO + mr, O - 1);
  const float bm = (tileO + mr < O) ? 1.0f : 0.0f;
  v8f acc = {0.f, 0.f, 0.f, 0.f, 0.f, 0.f, 0.f, 0.f};
  for (int k0 = 0; k0 < K; k0 += 4) {
    const int ka = k0 + (half << 1);
    const float m0 = (ka < K) ? 1.0f : 0.0f;
    const float m1 = (ka + 1 < K) ? 1.0f : 0.0f;
    const int k1 = min(ka, K - 1);
    const int k2 = min(ka + 1, K - 1);
    v2f a, b;
    a.x = X[(size_t)arow * xs + k1] * (am * m0);
    a.y = X[(size_t)arow * xs + k2] * (am * m1);
    b.x = W[(size_t)bcol * K + k1] * (bm * m0);
    b.y = W[(size_t)bcol * K + k2] * (bm * m1);
    acc = __builtin_amdgcn_wmma_f32_16x16x4_f32(false, a, false, b, (short)0, acc, false, false);
  }
#pragma unroll
  for (int v = 0; v < 8; ++v) {
    const int row = tileM + v + (half << 3);
    const int col = tileO + mr;
    if (row < M && col < O) {
      float val = acc[v];
      if (bias) val += bias[col];
      val = apply_act(val, act);
      if (res) val += res[(size_t)row * O + col];
      Y[(size_t)row * O + col] = val;
    }
  }
}

// LayerNorm over C=128 features; one token per 128-thread block.
__global__ void layernorm_k(const float* X, const float* g, const float* b,
                            float* Y, int C) {
  __shared__ float red[HD];
  const int t = blockIdx.x;
  const int c = threadIdx.x;
  const float v = X[(size_t)t * C + c];
  red[c] = v;
  __syncthreads();
  for (int s = C >> 1; s > 0; s >>= 1) {
    if (c < s) red[c] += red[c + s];
    __syncthreads();
  }
  const float mean = red[0] / (float)C;
  __syncthreads();
  const float d = v - mean;
  red[c] = d * d;
  __syncthreads();
  for (int s = C >> 1; s > 0; s >>= 1) {
    if (c < s) red[c] += red[c + s];
    __syncthreads();
  }
  const float var = red[0] / (float)C;
  const float inv = rsqrtf(var + 1e-5f);
  Y[(size_t)t * C + c] = d * inv * g[c] + b[c];
}

// Depthwise conv along L (per segment). w:[C,1,K]. Optional residual (same
// layout/stride as X) and activation. Output dense stride C.
__global__ void dwconv_k(const float* X, int xs, const float* w, const float* b,
                         const float* res, float* Y, int Nseg, int L, int C,
                         int K, int PL, int act) {
  const int idx = blockIdx.x * blockDim.x + threadIdx.x;
  const int total = Nseg * L * C;
  if (idx >= total) return;
  const int c = idx % C;
  const int l = (idx / C) % L;
  const int n = idx / (C * L);
  float acc = b[c];
  for (int j = 0; j < K; ++j) {
    const int ll = l + j - PL;
    if (ll >= 0 && ll < L) acc += w[c * K + j] * X[(size_t)(n * L + ll) * xs + c];
  }
  if (res) acc += res[(size_t)(n * L + l) * xs + c];
  Y[(size_t)(n * L + l) * C + c] = apply_act(acc, act);
}

// Selective scan: one block per segment, 256 threads = DI channels,
// 16 SSM states per channel in registers, B/C broadcast through LDS.
__global__ void scan_k(const float* u, const float* delta, const float* A_log,
                       const float* Dvec, const float* dbc, float* y, int L) {
  const int n = blockIdx.x;
  const int di = threadIdx.x;
  float A[DSC];
#pragma unroll
  for (int s = 0; s < DSC; ++s) A[s] = -expf(A_log[di * DSC + s]);
  const float Dv = Dvec[di];
  float h[DSC];
#pragma unroll
  for (int s = 0; s < DSC; ++s) h[s] = 0.0f;
  __shared__ float Bs[DSC];
  __shared__ float Cs[DSC];
  for (int l = 0; l < L; ++l) {
    const float* row = dbc + (size_t)(n * L + l) * 40;
    if (di < 32) {
      const float t = row[8 + di]; // cols 8..23 = B, 24..39 = C
      if (di < DSC) Bs[di] = t; else Cs[di - DSC] = t;
    }
    __syncthreads();
    const float d = delta[(size_t)(n * L + l) * DIC + di];
    const float uu = u[(size_t)(n * L + l) * DIC + di];
    float yy = 0.0f;
#pragma unroll
    for (int s = 0; s < DSC; ++s) {
      const float dA = expf(d * A[s]);
      h[s] = dA * h[s] + d * Bs[s] * uu;
      yy += h[s] * Cs[s];
    }
    y[(size_t)(n * L + l) * DIC + di] = yy + uu * Dv;
    __syncthreads();
  }
}

// y2 = y * silu(z), z = xz[:, 256+c]
__global__ void gate_k(const float* y, const float* xz, float* out, int Ntok) {
  const int idx = blockIdx.x * blockDim.x + threadIdx.x;
  if (idx >= Ntok * DIC) return;
  const int c = idx % DIC;
  const int t = idx / DIC;
  const float z = xz[(size_t)t * (2 * DIC) + DIC + c];
  out[idx] = y[idx] * (z / (1.0f + expf(-z)));
}

// Per-segment masked mean/max pool -> wv = 0.5*(mean+max); zero masked windows.
__global__ void segpool_k(const float* x, const int* seg_len, const int* seq_len,
                          float* seq, int L, int C) {
  const int sidx = blockIdx.x;
  const int c = threadIdx.x;
  const int len = seg_len[sidx];
  const int lc = len > 1 ? len : 1;
  float sum = 0.0f, mx = -INFINITY;
  for (int l = 0; l < len; ++l) {
    const float v = x[(size_t)(sidx * L + l) * C + c];
    sum += v;
    mx = fmaxf(mx, v);
  }
  const float wv = 0.5f * (sum / (float)lc + mx);
  const int b = sidx / SPC, sp = sidx % SPC;
  const float sm = (sp < seq_len[b]) ? 1.0f : 0.0f;
  seq[(size_t)sidx * C + c] = wv * sm;
}

// Masked mean over windows -> g[b,c]
__global__ void seqpool_k(const float* seq, const int* seq_len, float* g) {
  const int b = blockIdx.x;
  const int c = threadIdx.x;
  const int sl = seq_len[b];
  float sum = 0.0f;
  for (int sp = 0; sp < SPC; ++sp)
    if (sp < sl) sum += seq[(size_t)(b * SPC + sp) * HD + c];
  g[(size_t)b * HD + c] = sum / (float)(sl > 1 ? sl : 1);
}

// ---------------- host side ----------------
struct MambaP {
  const float *in_w, *conv_w, *conv_b, *x_w, *dt_w, *dt_b, *A_log, *D, *out_w;
};
struct BlockP {
  const float *ln1_g, *ln1_b, *conv_w, *conv_b;
  MambaP ssm;
  const float *ln2_g, *ln2_b, *mlp_w1, *mlp_b1, *mlp_w2, *mlp_b2;
};
struct NetP {
  const float *a_w1, *a_b1, *a_w2, *a_b2;
  BlockP intra[2], inter[2];
  const float *h_ln_g, *h_ln_b, *h_w1, *h_b1, *h_w2, *h_b2;
  const float* seg_inputs;
  const int *seg_len, *seq_len;
};

static inline void gemm(hipStream_t st, const float* X, int xs, const float* W,
                        const float* bias, const float* res, float* Y, int M,
                        int K, int O, int act) {
  if ((K % 4 == 0) && (M % 16 == 0) && (O % 64 == 0)) {
    dim3 grid(M / 16, O / 64);
    gemm_wmma_f32_fast<<<grid, 32, 0, st>>>(X, xs, W, bias, res, Y, K, O, act);
  } else {
    dim3 grid((M + 15) / 16, (O + 15) / 16);
    gemm_wmma_f32_gen<<<grid, 32, 0, st>>>(X, xs, W, bias, res, Y, M, K, O, act);
  }
}

static void run_block(hipStream_t st, const BlockP& p, float* x, int Nseg, int L,
                      float* bh, float* buin, float* bxz, float* bu, float* bdbc,
                      float* bdelta, float* by) {
  const int NTok = Nseg * L;
  layernorm_k<<<NTok, HD, 0, st>>>(x, p.ln1_g, p.ln1_b, bh, HD);
  {
    const int tot = NTok * HD; // hc + h (fused residual)
    dwconv_k<<<(tot + 255) / 256, 256, 0, st>>>(bh, HD, p.conv_w, p.conv_b, bh,
                                                buin, Nseg, L, HD, 3, 1, 0);
  }
  gemm(st, buin, HD, p.ssm.in_w, nullptr, nullptr, bxz, NTok, HD, 2 * DIC, 0);
  {
    const int tot = NTok * DIC; // causal k=4 conv on u-half of xz, then SiLU
    dwconv_k<<<(tot + 255) / 256, 256, 0, st>>>(bxz, 2 * DIC, p.ssm.conv_w,
                                                p.ssm.conv_b, nullptr, bu, Nseg,
                                                L, DIC, 4, 3, 2);
  }
  gemm(st, bu, DIC, p.ssm.x_w, nullptr, nullptr, bdbc, NTok, DIC, 40, 0);
  gemm(st, bdbc, 40, p.ssm.dt_w, p.ssm.dt_b, nullptr, bdelta, NTok, 8, DIC, 3);
  scan_k<<<Nseg, DIC, 0, st>>>(bu, bdelta, p.ssm.A_log, p.ssm.D, bdbc, by, L);
  {
    const int tot = NTok * DIC;
    gate_k<<<(tot + 255) / 256, 256, 0, st>>>(by, bxz, by, NTok);
  }
  gemm(st, by, DIC, p.ssm.out_w, nullptr, x, x, NTok, DIC, HD, 0); // +residual
  layernorm_k<<<NTok, HD, 0, st>>>(x, p.ln2_g, p.ln2_b, bh, HD);
  gemm(st, bh, HD, p.mlp_w1, p.mlp_b1, nullptr, bxz, NTok, HD, 4 * HD, 1);
  gemm(st, bxz, 4 * HD, p.mlp_w2, p.mlp_b2, x, x, NTok, 4 * HD, HD, 0);
}

extern "C" void kernel_launch(void* const* d_in, const int* in_sizes, int n_in,
                              void* d_out, int out_size, void* d_ws,
                              size_t ws_size, hipStream_t stream) {
  NetP P;
  int ii = 0;
  auto nxt = [&]() { return (const float*)d_in[ii++]; };
  const bool insertion = (in_sizes[0] == SEGS * LMAXC * FEATC);
  if (insertion) {
    P.seg_inputs = nxt();
    P.a_w1 = nxt(); P.a_b1 = nxt(); P.a_w2 = nxt(); P.a_b2 = nxt();
    BlockP* order[4] = {&P.intra[0], &P.intra[1], &P.inter[0], &P.inter[1]};
    for (int bi = 0; bi < 4; ++bi) {
      BlockP& b = *order[bi];
      b.ln1_g = nxt(); b.ln1_b = nxt(); b.conv_w = nxt(); b.conv_b = nxt();
      b.ssm.in_w = nxt(); b.ssm.conv_w = nxt(); b.ssm.conv_b = nxt();
      b.ssm.x_w = nxt(); b.ssm.dt_w = nxt(); b.ssm.dt_b = nxt();
      b.ssm.A_log = nxt(); b.ssm.D = nxt(); b.ssm.out_w = nxt();
      b.ln2_g = nxt(); b.ln2_b = nxt();
      b.mlp_w1 = nxt(); b.mlp_b1 = nxt(); b.mlp_w2 = nxt(); b.mlp_b2 = nxt();
    }
    P.h_ln_g = nxt(); P.h_ln_b = nxt(); P.h_w1 = nxt(); P.h_b1 = nxt();
    P.h_w2 = nxt(); P.h_b2 = nxt();
    P.seg_len = (const int*)d_in[ii++];
    P.seq_len = (const int*)d_in[ii++];
  } else {
    // JAX pytree alphabetical flattening
    P.a_b1 = nxt(); P.a_b2 = nxt(); P.a_w1 = nxt(); P.a_w2 = nxt();
    P.h_b1 = nxt(); P.h_b2 = nxt(); P.h_ln_b = nxt(); P.h_ln_g = nxt();
    P.h_w1 = nxt(); P.h_w2 = nxt();
    BlockP* order[4] = {&P.inter[0], &P.inter[1], &P.intra[0], &P.intra[1]};
    for (int bi = 0; bi < 4; ++bi) {
      BlockP& b = *order[bi];
      b.conv_b = nxt(); b.conv_w = nxt();
      b.ln1_b = nxt(); b.ln1_g = nxt(); b.ln2_b = nxt(); b.ln2_g = nxt();
      b.mlp_b1 = nxt(); b.mlp_b2 = nxt(); b.mlp_w1 = nxt(); b.mlp_w2 = nxt();
      b.ssm.A_log = nxt(); b.ssm.D = nxt(); b.ssm.conv_b = nxt();
      b.ssm.conv_w = nxt(); b.ssm.dt_b = nxt(); b.ssm.dt_w = nxt();
      b.ssm.in_w = nxt(); b.ssm.out_w = nxt(); b.ssm.x_w = nxt();
    }
    P.seg_inputs = nxt();
    P.seg_len = (const int*)d_in[ii++];
    P.seq_len = (const int*)d_in[ii++];
  }

  // workspace slabs
  float* w = (float*)d_ws;
  const size_t NT = (size_t)SEGS * LMAXC; // 16384 tokens (intra phase)
  float* bx = w;     w += NT * HD;
  float* bh = w;     w += NT * HD;
  float* buin = w;   w += NT * HD;
  float* bxz = w;    w += NT * 2 * DIC;
  float* bu = w;     w += NT * DIC;
  float* bdbc = w;   w += NT * 64; // 40 used, padded
  float* bdelta = w; w += NT * DIC;
  float* by = w;     w += NT * DIC;
  float* bseq = w;   w += (size_t)SEGS * HD;
  float* bg = w;     w += (size_t)BATCH * HD;
  float* bgn = w;    w += (size_t)BATCH * HD;
  float* bgh = w;    w += (size_t)BATCH * (HD / 2);
  (void)ws_size; (void)n_in; (void)out_size;

  const int NTok = (int)NT;
  // adapter: gelu(x@w1^T+b1)@w2^T+b2  (seg_inputs already masked by setup)
  gemm(stream, P.seg_inputs, FEATC, P.a_w1, P.a_b1, nullptr, bh, NTok, FEATC, HD, 1);
  gemm(stream, bh, HD, P.a_w2, P.a_b2, nullptr, bx, NTok, HD, HD, 0);

  // intra blocks over [64,256,128]
  run_block(stream, P.intra[0], bx, SEGS, LMAXC, bh, buin, bxz, bu, bdbc, bdelta, by);
  run_block(stream, P.intra[1], bx, SEGS, LMAXC, bh, buin, bxz, bu, bdbc, bdelta, by);

  // pool segments -> [8,8,128], zero masked windows
  segpool_k<<<SEGS, HD, 0, stream>>>(bx, P.seg_len, P.seq_len, bseq, LMAXC, HD);

  // inter blocks over [8,8,128]
  run_block(stream, P.inter[0], bseq, BATCH, SPC, bh, buin, bxz, bu, bdbc, bdelta, by);
  run_block(stream, P.inter[1], bseq, BATCH, SPC, bh, buin, bxz, bu, bdbc, bdelta, by);

  // masked mean over windows, head
  seqpool_k<<<BATCH, HD, 0, stream>>>(bseq, P.seq_len, bg);
  layernorm_k<<<BATCH, HD, 0, stream>>>(bg, P.h_ln_g, P.h_ln_b, bgn, HD);
  gemm(stream, bgn, HD, P.h_w1, P.h_b1, nullptr, bgh, BATCH, HD, HD / 2, 1);
  gemm(stream, bgh, HD / 2, P.h_w2, P.h_b2, nullptr, (float*)d_out, BATCH, HD / 2, NCC, 0);
}